// model_11184094839076
// MI455X (gfx1250) — compile-verified
//
#include <hip/hip_runtime.h>

// Brownian-bridge Euler-Maruyama scan, MI455X (gfx1250).
// N=2048 rows, D=128 dims, M=256 steps. Memory-bound streaming kernel:
//   - 1 wave32 per row, each lane owns a float4 of the D dimension
//   - noise stream prefetched via async global->LDS copies (double buffer)
//   - outputs written with non-temporal b128 stores (touched once, > L2)

typedef __attribute__((ext_vector_type(4))) float v4f;

#define N_TRAJ 2048
#define D_DIM  128
#define M_STEPS 256
#define CHUNK  8                         // steps per async chunk
#define NCHUNKS (M_STEPS / CHUNK)        // 32
#define WAVES_PER_BLOCK 4
#define THREADS (WAVES_PER_BLOCK * 32)

#define H_CONST   0.00390625f            // 1/256
#define C_CONST   0.0625f                // sigma*sqrt(h) = 1/16
#define SBK_CONST (-16.0f)               // -inv_sigma_h*sigma*sqrt_h = -16

__global__ __launch_bounds__(THREADS)
void bridge_scan_kernel(const float* __restrict__ init,
                        const float* __restrict__ term,
                        const float* __restrict__ noise,
                        float* __restrict__ traj,
                        float* __restrict__ sb)
{
    // 4 waves * 2 buffers * 8 rows * 128 floats = 32 KB
    __shared__ __align__(16) float smem[WAVES_PER_BLOCK * 2 * CHUNK * D_DIM];

    const int wave = threadIdx.x >> 5;
    const int lane = threadIdx.x & 31;
    const int n    = blockIdx.x * WAVES_PER_BLOCK + wave;
    const int d0   = lane * 4;

    float*       wbase = &smem[wave * (2 * CHUNK * D_DIM)];
    const float* nrow  = noise + (size_t)n * M_STEPS * D_DIM + d0;
    float*       trow  = traj  + (size_t)n * (M_STEPS + 1) * D_DIM + d0;
    float*       srow  = sb    + (size_t)n * M_STEPS * D_DIM + d0;

    // Issue one chunk (8 rows x 512B) of async global->LDS copies.
    // VDST VGPR = LDS byte offset (low 32 bits of generic LDS address).
    auto issue_chunk = [&](int c) {
        float*       ldst = wbase + (c & 1) * (CHUNK * D_DIM) + d0;
        const float* gsrc = nrow + (size_t)c * CHUNK * D_DIM;
#pragma unroll
        for (int r = 0; r < CHUNK; ++r) {
            unsigned int ldsoff = (unsigned int)(size_t)(ldst + r * D_DIM);
            const float* ga     = gsrc + (size_t)r * D_DIM;
            asm volatile("global_load_async_to_lds_b128 %0, %1, off"
                         :: "v"(ldsoff), "v"(ga)
                         : "memory");
        }
    };

    // Load state; trajectories[:,0,:] = initial_states (bitwise).
    v4f x = *(const v4f*)(init + (size_t)n * D_DIM + d0);
    const v4f z = *(const v4f*)(term + (size_t)n * D_DIM + d0);
    __builtin_nontemporal_store(x, (v4f*)trow);

    issue_chunk(0);
    issue_chunk(1);

    for (int ch = 0; ch < NCHUNKS; ++ch) {
        // Ensure current chunk's 8 async loads have landed in LDS.
        if (ch < NCHUNKS - 2)
            asm volatile("s_wait_asynccnt 0x8" ::: "memory");
        else
            asm volatile("s_wait_asynccnt 0x0" ::: "memory");

        const float* lsrc = wbase + (ch & 1) * (CHUNK * D_DIM) + d0;

#pragma unroll
        for (int r = 0; r < CHUNK; ++r) {
            const int m = ch * CHUNK + r;
            const v4f e = *(const v4f*)(lsrc + r * D_DIM);  // ds_load_b128

            // a_m = h/(T - t_m) = 1/(256 - m)
            const float a = 1.0f / (float)(M_STEPS - m);

            x.x = fmaf(C_CONST, e.x, fmaf(a, z.x - x.x, x.x));
            x.y = fmaf(C_CONST, e.y, fmaf(a, z.y - x.y, x.y));
            x.z = fmaf(C_CONST, e.z, fmaf(a, z.z - x.z, x.z));
            x.w = fmaf(C_CONST, e.w, fmaf(a, z.w - x.w, x.w));
            __builtin_nontemporal_store(x, (v4f*)(trow + (size_t)(m + 1) * D_DIM));

            v4f s4;
            s4.x = SBK_CONST * e.x;
            s4.y = SBK_CONST * e.y;
            s4.z = SBK_CONST * e.z;
            s4.w = SBK_CONST * e.w;
            __builtin_nontemporal_store(s4, (v4f*)(srow + (size_t)m * D_DIM));
        }

        // Buffer (ch & 1) is now fully consumed into VGPRs; refill it.
        if (ch + 2 < NCHUNKS)
            issue_chunk(ch + 2);
    }
}

extern "C" void kernel_launch(void* const* d_in, const int* in_sizes, int n_in,
                              void* d_out, int out_size, void* d_ws, size_t ws_size,
                              hipStream_t stream)
{
    const float* init  = (const float*)d_in[0];   // (2048, 128)
    const float* term  = (const float*)d_in[1];   // (2048, 128)
    const float* noise = (const float*)d_in[2];   // (2048, 256, 128)

    float* traj = (float*)d_out;                                   // (2048, 257, 128)
    float* sb   = traj + (size_t)N_TRAJ * (M_STEPS + 1) * D_DIM;   // (2048, 256, 128)

    dim3 grid(N_TRAJ / WAVES_PER_BLOCK);
    bridge_scan_kernel<<<grid, THREADS, 0, stream>>>(init, term, noise, traj, sb);
}